// IsometryInverse_46497315946833
// MI455X (gfx1250) — compile-verified
//
#include <hip/hip_runtime.h>

// Batched 4x4 fp32 matrix inverse — memory-bandwidth-bound on MI455X
// (268 MB streamed @ 23.3 TB/s ≈ 11.5 us floor).
// Pipeline: async global->LDS staging (coalesced b128, NT policy since the
// stream exceeds the 192 MB L2), per-thread 2x2-subdeterminant inverse
// (~110 VALU vs ~220 for the full cofactor form), async LDS->global drain.

typedef float v4f __attribute__((ext_vector_type(4)));
typedef int   v4i __attribute__((ext_vector_type(4)));

#define MATS_PER_BLOCK 256
#define SM_STRIDE 20   // floats per LDS matrix slot: 16 + 4 pad (<=2-way bank conflict, keeps 16B align)

#if defined(__gfx1250__) && \
    __has_builtin(__builtin_amdgcn_global_load_async_to_lds_b128) && \
    __has_builtin(__builtin_amdgcn_global_store_async_from_lds_b128)
#define USE_ASYNC_LDS 1
#define CPOL_NT 1   // TH[2:0] = 1 -> non-temporal (zero-reuse 268 MB stream)
typedef __attribute__((address_space(1))) v4i v4i_g;   // global-AS int4
typedef __attribute__((address_space(3))) v4i v4i_l;   // LDS-AS int4
#else
#define USE_ASYNC_LDS 0
#endif

#if defined(__gfx1250__) && __has_builtin(__builtin_amdgcn_s_wait_asynccnt)
#define WAIT_ASYNC() __builtin_amdgcn_s_wait_asynccnt(0)
#elif USE_ASYNC_LDS
#define WAIT_ASYNC() asm volatile("s_wait_asynccnt 0" ::: "memory")
#else
#define WAIT_ASYNC() do {} while (0)
#endif

// 4x4 inverse via 2x2 sub-determinants (Laplace block expansion).
// Row-major m[r*4+c]; works for batched general matrices.
__device__ __forceinline__ void invert4x4(const float* __restrict__ m,
                                          float* __restrict__ b) {
    const float s0 = m[0] * m[5]  - m[4] * m[1];
    const float s1 = m[0] * m[6]  - m[4] * m[2];
    const float s2 = m[0] * m[7]  - m[4] * m[3];
    const float s3 = m[1] * m[6]  - m[5] * m[2];
    const float s4 = m[1] * m[7]  - m[5] * m[3];
    const float s5 = m[2] * m[7]  - m[6] * m[3];

    const float c5 = m[10] * m[15] - m[14] * m[11];
    const float c4 = m[9]  * m[15] - m[13] * m[11];
    const float c3 = m[9]  * m[14] - m[13] * m[10];
    const float c2 = m[8]  * m[15] - m[12] * m[11];
    const float c1 = m[8]  * m[14] - m[12] * m[10];
    const float c0 = m[8]  * m[13] - m[12] * m[9];

    const float det = s0 * c5 - s1 * c4 + s2 * c3 + s3 * c2 - s4 * c1 + s5 * c0;
    const float r   = 1.0f / det;

    b[0]  = ( m[5]  * c5 - m[6]  * c4 + m[7]  * c3) * r;
    b[1]  = (-m[1]  * c5 + m[2]  * c4 - m[3]  * c3) * r;
    b[2]  = ( m[13] * s5 - m[14] * s4 + m[15] * s3) * r;
    b[3]  = (-m[9]  * s5 + m[10] * s4 - m[11] * s3) * r;

    b[4]  = (-m[4]  * c5 + m[6]  * c2 - m[7]  * c1) * r;
    b[5]  = ( m[0]  * c5 - m[2]  * c2 + m[3]  * c1) * r;
    b[6]  = (-m[12] * s5 + m[14] * s2 - m[15] * s1) * r;
    b[7]  = ( m[8]  * s5 - m[10] * s2 + m[11] * s1) * r;

    b[8]  = ( m[4]  * c4 - m[5]  * c2 + m[7]  * c0) * r;
    b[9]  = (-m[0]  * c4 + m[1]  * c2 - m[3]  * c0) * r;
    b[10] = ( m[12] * s4 - m[13] * s2 + m[15] * s0) * r;
    b[11] = (-m[8]  * s4 + m[9]  * s2 - m[11] * s0) * r;

    b[12] = (-m[4]  * c3 + m[5]  * c1 - m[6]  * c0) * r;
    b[13] = ( m[0]  * c3 - m[1]  * c1 + m[2]  * c0) * r;
    b[14] = (-m[12] * s3 + m[13] * s1 - m[14] * s0) * r;
    b[15] = ( m[8]  * s3 - m[9]  * s1 + m[10] * s0) * r;
}

// ---- staging helpers ----
__device__ __forceinline__ void stage_in_one(const v4f* gin, float* lds, int f) {
    const int mi = f >> 2, c = f & 3;
    float* dst = &lds[mi * SM_STRIDE + c * 4];
#if USE_ASYNC_LDS
    __builtin_amdgcn_global_load_async_to_lds_b128(
        (v4i_g*)(gin + f), (v4i_l*)dst, /*offset=*/0, /*cpol=*/CPOL_NT);
#else
    *reinterpret_cast<v4f*>(dst) = __builtin_nontemporal_load(gin + f);
#endif
}

__device__ __forceinline__ void stage_out_one(v4f* gout, float* lds, int f) {
    const int mi = f >> 2, c = f & 3;
    float* src = &lds[mi * SM_STRIDE + c * 4];
#if USE_ASYNC_LDS
    __builtin_amdgcn_global_store_async_from_lds_b128(
        (v4i_g*)(gout + f), (v4i_l*)src, /*offset=*/0, /*cpol=*/CPOL_NT);
#else
    __builtin_nontemporal_store(*reinterpret_cast<const v4f*>(src), gout + f);
#endif
}

__device__ __forceinline__ void compute_slot(float* lds, int t) {
    float m[16], o[16];
    float* slot = &lds[t * SM_STRIDE];
#pragma unroll
    for (int c = 0; c < 4; ++c)
        *reinterpret_cast<v4f*>(&m[4 * c]) = *reinterpret_cast<const v4f*>(&slot[4 * c]);
    invert4x4(m, o);
#pragma unroll
    for (int c = 0; c < 4; ++c)
        *reinterpret_cast<v4f*>(&slot[4 * c]) = *reinterpret_cast<const v4f*>(&o[4 * c]);
}

__global__ __launch_bounds__(MATS_PER_BLOCK)
void IsometryInverse_kernel(const float* __restrict__ in,
                            float* __restrict__ out, int nmat) {
    __shared__ __align__(16) float lds[MATS_PER_BLOCK * SM_STRIDE];  // 20 KB

    const int t     = threadIdx.x;
    const int mbase = blockIdx.x * MATS_PER_BLOCK;
    const int mrem  = nmat - mbase;                        // matrices this block

    const v4f* __restrict__ gin  = reinterpret_cast<const v4f*>(in)  + (size_t)mbase * 4;
    v4f* __restrict__       gout = reinterpret_cast<v4f*>(out)       + (size_t)mbase * 4;

    if (mrem >= MATS_PER_BLOCK) {
        // ---------- fast path: full tile, branch-free ----------
#pragma unroll
        for (int j = 0; j < 4; ++j)
            stage_in_one(gin, lds, t + j * MATS_PER_BLOCK);
        WAIT_ASYNC();
        __syncthreads();

        compute_slot(lds, t);
        __syncthreads();

#pragma unroll
        for (int j = 0; j < 4; ++j)
            stage_out_one(gout, lds, t + j * MATS_PER_BLOCK);
        WAIT_ASYNC();   // async stores read LDS; drain before wave teardown
    } else {
        // ---------- tail path: guarded ----------
        const int nv4 = mrem * 4;
#pragma unroll
        for (int j = 0; j < 4; ++j) {
            const int f = t + j * MATS_PER_BLOCK;
            if (f < nv4) stage_in_one(gin, lds, f);
        }
        WAIT_ASYNC();
        __syncthreads();

        if (t < mrem) compute_slot(lds, t);
        __syncthreads();

#pragma unroll
        for (int j = 0; j < 4; ++j) {
            const int f = t + j * MATS_PER_BLOCK;
            if (f < nv4) stage_out_one(gout, lds, f);
        }
        WAIT_ASYNC();
    }
}

extern "C" void kernel_launch(void* const* d_in, const int* in_sizes, int n_in,
                              void* d_out, int out_size, void* d_ws, size_t ws_size,
                              hipStream_t stream) {
    const float* in  = (const float*)d_in[0];
    float*       out = (float*)d_out;
    const int nmat   = in_sizes[0] / 16;              // 2,097,152
    const int blocks = (nmat + MATS_PER_BLOCK - 1) / MATS_PER_BLOCK;
    IsometryInverse_kernel<<<blocks, MATS_PER_BLOCK, 0, stream>>>(in, out, nmat);
}